// PlainGCN_57071525429591
// MI455X (gfx1250) — compile-verified
//
#include <hip/hip_runtime.h>
#include <hip/hip_bf16.h>

typedef __attribute__((ext_vector_type(2))) float v2f;
typedef __attribute__((ext_vector_type(8))) float v8f;

#define H_DIM 256          // hidden width of every GCN layer
#define NT 4               // 16-wide N-tiles per wave (4 waves * 4 tiles * 16 = 256 cols)

// ---------------------------------------------------------------------------
// degree / normalization
// ---------------------------------------------------------------------------
__global__ void k_init_deg(float* __restrict__ deg, int n) {
  int i = blockIdx.x * blockDim.x + threadIdx.x;
  if (i < n) deg[i] = 1.0f;                       // self-loop contributes 1
}

__global__ void k_count_deg(const int* __restrict__ col, float* __restrict__ deg, int e) {
  int i = blockIdx.x * blockDim.x + threadIdx.x;
  if (i < e) atomicAdd(&deg[col[i]], 1.0f);
}

__global__ void k_dinv(float* __restrict__ deg, int n) {
  int i = blockIdx.x * blockDim.x + threadIdx.x;
  if (i < n) {
    float d = deg[i];
    deg[i] = (d > 0.0f) ? rsqrtf(d) : 0.0f;       // deg >= 1 always (self loops)
  }
}

// ---------------------------------------------------------------------------
// dense GEMM  h[M,256] = X[M,F] @ W[F,256]  via V_WMMA_F32_16X16X4_F32
// grid.x = M/16, block = 128 (4 waves); wave w owns columns [w*64, w*64+64)
// A frag (16x4 f32): lanes 0-15 hold K=k,k+1 ; lanes 16-31 hold K=k+2,k+3 ; M = lane&15
// B frag (4x16 f32): symmetric striping, N = lane&15
// C/D  (16x16 f32): VGPR i -> M = i + 8*(lane>>4), N = lane&15
// ---------------------------------------------------------------------------
__global__ __launch_bounds__(128)
void k_gemm_wmma(const float* __restrict__ X, const float* __restrict__ W,
                 float* __restrict__ Hout, int F) {
  const int lane  = threadIdx.x & 31;
  const int wave  = threadIdx.x >> 5;
  const int hi    = lane >> 4;        // 0: low half-wave, 1: high half-wave
  const int ln    = lane & 15;
  const int mbase = blockIdx.x << 4;
  const int nbase = wave * (NT * 16);

  v8f acc[NT] = {};

  const float* xrow = X + (size_t)(mbase + ln) * (size_t)F;

  for (int k = 0; k < F; k += 4) {
    // A fragment: two consecutive K values per lane -> one float2 load
    v2f a = *(const v2f*)(xrow + k + 2 * hi);
#pragma unroll
    for (int t = 0; t < NT; ++t) {
      const int n = nbase + t * 16 + ln;
      v2f b;
      b.x = W[(size_t)(k + 2 * hi)     * H_DIM + n];
      b.y = W[(size_t)(k + 2 * hi + 1) * H_DIM + n];
      acc[t] = __builtin_amdgcn_wmma_f32_16x16x4_f32(
          /*neg_a=*/false, a, /*neg_b=*/false, b,
          /*c_mod=*/(short)0, acc[t], /*reuse_a=*/false, /*reuse_b=*/false);
    }
  }

#pragma unroll
  for (int t = 0; t < NT; ++t) {
    const int n = nbase + t * 16 + ln;
#pragma unroll
    for (int i = 0; i < 8; ++i) {
      Hout[(size_t)(mbase + i + 8 * hi) * H_DIM + n] = acc[t][i];
    }
  }
}

// ---------------------------------------------------------------------------
// aggregation:  agg[i] = dinv[i]^2 * h[i]   (self loop, initializes agg)
//               agg[c] += dinv[r]*dinv[c] * h[r]   (per edge, atomic)
//               act[i] = relu(agg[i] + bias)       (in place)
// ---------------------------------------------------------------------------
__global__ void k_selfloop(const float* __restrict__ h, const float* __restrict__ dinv,
                           float* __restrict__ agg) {
  const int i = blockIdx.x;
  const int f = threadIdx.x;
  const float w = dinv[i] * dinv[i];
  agg[(size_t)i * H_DIM + f] = w * h[(size_t)i * H_DIM + f];
}

__global__ void k_edge_scatter(const int* __restrict__ row, const int* __restrict__ col,
                               const float* __restrict__ h, const float* __restrict__ dinv,
                               float* __restrict__ agg) {
  const int e = blockIdx.x;
  const int f = threadIdx.x;
  const int r = row[e];
  const int c = col[e];
  const float w = dinv[r] * dinv[c];
  atomicAdd(&agg[(size_t)c * H_DIM + f], w * h[(size_t)r * H_DIM + f]);
}

__global__ void k_bias_relu(float* __restrict__ a, const float* __restrict__ b) {
  const int i = blockIdx.x;
  const int f = threadIdx.x;
  const size_t idx = (size_t)i * H_DIM + f;
  const float v = a[idx] + b[f];
  a[idx] = v > 0.0f ? v : 0.0f;
}

// ---------------------------------------------------------------------------
// global mean pool + final linear
// ---------------------------------------------------------------------------
__global__ void k_zero_pool(float* __restrict__ pooled, float* __restrict__ cnt, int g) {
  int i = blockIdx.x * blockDim.x + threadIdx.x;
  if (i < g * H_DIM) pooled[i] = 0.0f;
  if (i < g) cnt[i] = 0.0f;
}

__global__ void k_pool_accum(const float* __restrict__ x, const int* __restrict__ batch,
                             float* __restrict__ pooled, float* __restrict__ cnt) {
  const int i = blockIdx.x;
  const int f = threadIdx.x;
  const int g = batch[i];
  atomicAdd(&pooled[(size_t)g * H_DIM + f], x[(size_t)i * H_DIM + f]);
  if (f == 0) atomicAdd(&cnt[g], 1.0f);
}

__global__ void k_final_linear(const float* __restrict__ pooled, const float* __restrict__ cnt,
                               const float* __restrict__ lw, const float* __restrict__ lb,
                               float* __restrict__ out, int g, int c) {
  const int t = blockIdx.x * blockDim.x + threadIdx.x;
  if (t >= g * c) return;
  const int gi = t / c;
  const int ci = t % c;
  const float inv = 1.0f / fmaxf(cnt[gi], 1.0f);
  float s = 0.0f;
  for (int f = 0; f < H_DIM; ++f)
    s += pooled[(size_t)gi * H_DIM + f] * lw[(size_t)f * c + ci];
  out[t] = s * 1.0f + lb[ci];  // note: mean applied below
}

__global__ void k_final_linear_mean(const float* __restrict__ pooled, const float* __restrict__ cnt,
                                    const float* __restrict__ lw, const float* __restrict__ lb,
                                    float* __restrict__ out, int g, int c) {
  const int t = blockIdx.x * blockDim.x + threadIdx.x;
  if (t >= g * c) return;
  const int gi = t / c;
  const int ci = t % c;
  const float inv = 1.0f / fmaxf(cnt[gi], 1.0f);
  float s = 0.0f;
  for (int f = 0; f < H_DIM; ++f)
    s += (pooled[(size_t)gi * H_DIM + f] * inv) * lw[(size_t)f * c + ci];
  out[t] = s + lb[ci];
}

// ---------------------------------------------------------------------------
// driver
// ---------------------------------------------------------------------------
extern "C" void kernel_launch(void* const* d_in, const int* in_sizes, int n_in,
                              void* d_out, int out_size, void* d_ws, size_t ws_size,
                              hipStream_t stream) {
  // setup_inputs order: x, edge_index, batch, W0,b0, W1,b1, W2,b2, lin_W, lin_b
  const float* x_in  = (const float*)d_in[0];
  const int*   eidx  = (const int*)d_in[1];
  const int*   batch = (const int*)d_in[2];
  const float* W0 = (const float*)d_in[3];  const float* b0 = (const float*)d_in[4];
  const float* W1 = (const float*)d_in[5];  const float* b1 = (const float*)d_in[6];
  const float* W2 = (const float*)d_in[7];  const float* b2 = (const float*)d_in[8];
  const float* lw = (const float*)d_in[9];  const float* lb = (const float*)d_in[10];
  float* out = (float*)d_out;

  const int F0 = 128;
  const int N  = in_sizes[0] / F0;          // 50000
  const int E  = in_sizes[1] / 2;           // 800000
  const int C  = in_sizes[10];              // 10
  const int G  = out_size / C;              // 64 graphs

  const int* row = eidx;
  const int* col = eidx + E;

  // workspace layout (floats)
  float* ws     = (float*)d_ws;
  float* dinv   = ws;                                   // N   (deg -> dinv in place)
  float* hbuf   = dinv + (size_t)N;                     // N*256
  float* bufA   = hbuf + (size_t)N * H_DIM;             // N*256
  float* bufB   = bufA + (size_t)N * H_DIM;             // N*256
  float* pooled = bufB + (size_t)N * H_DIM;             // G*256
  float* cnt    = pooled + (size_t)G * H_DIM;           // G

  const int TPB = 256;

  // normalization coefficients
  k_init_deg <<<(N + TPB - 1) / TPB, TPB, 0, stream>>>(dinv, N);
  k_count_deg<<<(E + TPB - 1) / TPB, TPB, 0, stream>>>(col, dinv, E);
  k_dinv     <<<(N + TPB - 1) / TPB, TPB, 0, stream>>>(dinv, N);

  const int mtiles = N / 16;  // 3125, N is a multiple of 16

  // ---- layer 0 : x_in[ ,128] -> bufA ----
  k_gemm_wmma   <<<mtiles, 128, 0, stream>>>(x_in, W0, hbuf, F0);
  k_selfloop    <<<N, H_DIM, 0, stream>>>(hbuf, dinv, bufA);
  k_edge_scatter<<<E, H_DIM, 0, stream>>>(row, col, hbuf, dinv, bufA);
  k_bias_relu   <<<N, H_DIM, 0, stream>>>(bufA, b0);

  // ---- layer 1 : bufA -> bufB ----
  k_gemm_wmma   <<<mtiles, 128, 0, stream>>>(bufA, W1, hbuf, H_DIM);
  k_selfloop    <<<N, H_DIM, 0, stream>>>(hbuf, dinv, bufB);
  k_edge_scatter<<<E, H_DIM, 0, stream>>>(row, col, hbuf, dinv, bufB);
  k_bias_relu   <<<N, H_DIM, 0, stream>>>(bufB, b1);

  // ---- layer 2 : bufB -> bufA ----
  k_gemm_wmma   <<<mtiles, 128, 0, stream>>>(bufB, W2, hbuf, H_DIM);
  k_selfloop    <<<N, H_DIM, 0, stream>>>(hbuf, dinv, bufA);
  k_edge_scatter<<<E, H_DIM, 0, stream>>>(row, col, hbuf, dinv, bufA);
  k_bias_relu   <<<N, H_DIM, 0, stream>>>(bufA, b2);

  // ---- global mean pool + classifier ----
  k_zero_pool   <<<(G * H_DIM + TPB - 1) / TPB, TPB, 0, stream>>>(pooled, cnt, G);
  k_pool_accum  <<<N, H_DIM, 0, stream>>>(bufA, batch, pooled, cnt);
  k_final_linear_mean<<<(G * C + 63) / 64, 64, 0, stream>>>(pooled, cnt, lw, lb, out, G, C);
}